// MbPAMLP2Layers_66039417143410
// MI455X (gfx1250) — compile-verified
//
#include <hip/hip_runtime.h>
#include <hip/hip_bf16.h>
#include <math.h>

// ---------------- problem constants ----------------
#define IDIM   512
#define NCLS   100
#define NPAD   112          // NCLS padded to 7 tiles of 16
#define CAP    100000
#define BSZ    128
#define KSEL   128
#define RSEL   (BSZ*KSEL)   // 16384 selected memories
#define STEPS  41
#define NCHUNK 16           // K-split for grad GEMM

typedef float v2f __attribute__((ext_vector_type(2)));
typedef float v8f __attribute__((ext_vector_type(8)));

__device__ __forceinline__ v8f wmma4(v2f a, v2f b, v8f c) {
  // D = A(16x4,f32) x B(4x16,f32) + C(16x16,f32), one wave
  return __builtin_amdgcn_wmma_f32_16x16x4_f32(false, a, false, b, (short)0, c,
                                               false, false);
}

// ============ emb = relu(input @ w1^T + b1) : [128,512] ============
// 8 m-tiles x 32 n-tiles = 256 waves
__global__ void k_emb(const float* __restrict__ in, const float* __restrict__ w1,
                      const float* __restrict__ b1, float* __restrict__ emb) {
  int wave = (blockIdx.x * blockDim.x + threadIdx.x) >> 5;
  int lane = threadIdx.x & 31;
  int mt = wave >> 5, nt = wave & 31;
  int half = lane >> 4, lr = lane & 15;
  const float* arow = in + (size_t)(mt * 16 + lr) * IDIM;
  const float* brow = w1 + (size_t)(nt * 16 + lr) * IDIM;
  v8f acc = {0.f,0.f,0.f,0.f,0.f,0.f,0.f,0.f};
#pragma unroll 8
  for (int k = 0; k < IDIM; k += 4) {
    v2f a = *(const v2f*)(arow + k + 2 * half);
    v2f b = *(const v2f*)(brow + k + 2 * half);
    acc = wmma4(a, b, acc);
  }
  int n = nt * 16 + lr;
  float bias = b1[n];
#pragma unroll
  for (int r = 0; r < 8; ++r) {
    int m = mt * 16 + r + 8 * half;
    float v = acc[r] + bias;
    emb[(size_t)m * IDIM + n] = v > 0.f ? v : 0.f;
  }
}

// ============ row squared-norms (one wave per row) ============
__global__ void k_rownorm(const float* __restrict__ x, float* __restrict__ out,
                          int nrows) {
  int wave = (blockIdx.x * blockDim.x + threadIdx.x) >> 5;
  int lane = threadIdx.x & 31;
  if (wave >= nrows) return;
  const float* row = x + (size_t)wave * IDIM;
  float s = 0.f;
#pragma unroll
  for (int i = lane * 4; i < IDIM; i += 128) {
    float4 v = *(const float4*)(row + i);
    s += v.x * v.x + v.y * v.y + v.z * v.z + v.w * v.w;
  }
  for (int m = 16; m; m >>= 1) s += __shfl_xor(s, m, 32);
  if (lane == 0) out[wave] = s;
}

// ============ kernT[b][m] = 1/(eps + ||mem_m||^2 + ||emb_b||^2 - 2 dot) ======
// One wave per m-tile (6250 waves); sweeps ALL 8 batch n-tiles in registers so
// mems_x (205 MB) is streamed from HBM exactly once.
__global__ void k_dist(const float* __restrict__ mem, const float* __restrict__ emb,
                       const float* __restrict__ memn, const float* __restrict__ embn,
                       float* __restrict__ kernT) {
  int wave = (blockIdx.x * blockDim.x + threadIdx.x) >> 5;
  int lane = threadIdx.x & 31;
  int mt = wave;
  if (mt >= CAP / 16) return;
  int half = lane >> 4, lr = lane & 15;
  const float* arow = mem + (size_t)(mt * 16 + lr) * IDIM;
  const float* brow = emb + (size_t)lr * IDIM;   // + nt*16*IDIM per n-tile
  v8f acc[8];
#pragma unroll
  for (int nt = 0; nt < 8; ++nt) acc[nt] = (v8f){0.f,0.f,0.f,0.f,0.f,0.f,0.f,0.f};
  for (int k = 0; k < IDIM; k += 4) {
    v2f a = *(const v2f*)(arow + k + 2 * half);
#pragma unroll
    for (int nt = 0; nt < 8; ++nt) {
      v2f b = *(const v2f*)(brow + (size_t)nt * 16 * IDIM + k + 2 * half);
      acc[nt] = wmma4(a, b, acc[nt]);
    }
  }
#pragma unroll
  for (int nt = 0; nt < 8; ++nt) {
    int n = nt * 16 + lr;
    float en = embn[n];
#pragma unroll
    for (int r = 0; r < 8; ++r) {
      int m = mt * 16 + r + 8 * half;
      float dsq = memn[m] + en - 2.f * acc[nt][r];
      kernT[(size_t)n * CAP + m] = 1.f / (1e-4f + dsq);
    }
  }
}

// ============ top-128 of 100000 per batch row (deterministic) ============
__global__ void k_topk(const float* __restrict__ kernT, int* __restrict__ idx,
                       float* __restrict__ wsel) {
  const int T = 512;
  int b = blockIdx.x, tid = threadIdx.x;
  const float* row = kernT + (size_t)b * CAP;
  __shared__ int red[512];
  __shared__ int sg[512], se[512];
  __shared__ int sidx[KSEL];
  __shared__ float svals[KSEL];
  __shared__ float fsum[KSEL];

  // MSB-greedy threshold: largest thr with count(u >= thr) >= KSEL
  unsigned prefix = 0u;
  for (int bit = 30; bit >= 0; --bit) {
    unsigned cand = prefix | (1u << bit);
    int c = 0;
    for (int i = tid; i < CAP; i += T) c += (__float_as_uint(row[i]) >= cand);
    red[tid] = c;
    __syncthreads();
    for (int s = T / 2; s; s >>= 1) {
      if (tid < s) red[tid] += red[tid + s];
      __syncthreads();
    }
    if (red[0] >= KSEL) prefix = cand;
    __syncthreads();
  }
  unsigned thr = prefix;

  // per-thread counts of strictly-greater and equal
  int cg = 0, ce = 0;
  for (int i = tid; i < CAP; i += T) {
    unsigned u = __float_as_uint(row[i]);
    cg += (u > thr);
    ce += (u == thr);
  }
  sg[tid] = cg; se[tid] = ce;
  __syncthreads();
  // inclusive Hillis-Steele scans (deterministic)
  for (int off = 1; off < T; off <<= 1) {
    int ag = (tid >= off) ? sg[tid - off] : 0;
    int ae = (tid >= off) ? se[tid - off] : 0;
    __syncthreads();
    sg[tid] += ag; se[tid] += ae;
    __syncthreads();
  }
  int totG = sg[T - 1];
  int pg = sg[tid] - cg;              // exclusive base for greater
  int pe = totG + (se[tid] - ce);     // ties start after all greater
  for (int i = tid; i < CAP; i += T) {
    float v = row[i];
    unsigned u = __float_as_uint(v);
    if (u > thr) {
      if (pg < KSEL) { sidx[pg] = i; svals[pg] = v; }
      pg++;
    } else if (u == thr) {
      if (pe < KSEL) { sidx[pe] = i; svals[pe] = v; }
      pe++;
    }
  }
  __syncthreads();
  if (tid < KSEL) fsum[tid] = svals[tid];
  __syncthreads();
  for (int s = KSEL / 2; s; s >>= 1) {
    if (tid < s) fsum[tid] += fsum[tid + s];
    __syncthreads();
  }
  // fold softmax-weight normalization and 1/(2*BSZ) loss scaling in here
  float inv = 1.f / (fsum[0] * (2.f * (float)BSZ));
  if (tid < KSEL) {
    idx[b * KSEL + tid]  = sidx[tid];
    wsel[b * KSEL + tid] = svals[tid] * inv;
  }
}

// ============ gather selected memory rows + labels (+ transposed copy) ======
__global__ void k_gather(const float* __restrict__ mem, const int* __restrict__ memy,
                         const int* __restrict__ idx, float* __restrict__ Msel,
                         float* __restrict__ MselT, int* __restrict__ lbl) {
  int wave = (blockIdx.x * blockDim.x + threadIdx.x) >> 5;
  int lane = threadIdx.x & 31;
  int src = idx[wave];
  const float4* s = (const float4*)(mem + (size_t)src * IDIM);
  float4* d = (float4*)(Msel + (size_t)wave * IDIM);
#pragma unroll
  for (int i = lane; i < IDIM / 4; i += 32) {
    float4 v = s[i];
    d[i] = v;
    MselT[(size_t)(4 * i + 0) * RSEL + wave] = v.x;
    MselT[(size_t)(4 * i + 1) * RSEL + wave] = v.y;
    MselT[(size_t)(4 * i + 2) * RSEL + wave] = v.z;
    MselT[(size_t)(4 * i + 3) * RSEL + wave] = v.w;
  }
  if (lane == 0) lbl[wave] = memy[src];
}

// ============ init tester params / Adam state ============
__global__ void k_initp(const float* __restrict__ w2, const float* __restrict__ b2,
                        float* __restrict__ w2c, float* __restrict__ mw,
                        float* __restrict__ vw, float* __restrict__ b2c,
                        float* __restrict__ mb, float* __restrict__ vb) {
  int i = blockIdx.x * blockDim.x + threadIdx.x;
  if (i < NCLS * IDIM) { w2c[i] = w2[i]; mw[i] = 0.f; vw[i] = 0.f; }
  if (i < NCLS)        { b2c[i] = b2[i]; mb[i] = 0.f; vb[i] = 0.f; }
}

// ============ param-distance coefficients ============
// coef = d/(2*alpha)/norm with d = nw2 + nb2 + 2e-12 (w1/b1 never move)
__global__ void k_norms(const float* __restrict__ w2c, const float* __restrict__ w2,
                        const float* __restrict__ b2c, const float* __restrict__ b2,
                        float* __restrict__ coef) {
  __shared__ float sw[256], sb[256];
  int tid = threadIdx.x;
  float aw = 0.f, ab = 0.f;
  for (int i = tid; i < NCLS * IDIM; i += 256) { float d = w2c[i] - w2[i]; aw += d * d; }
  for (int i = tid; i < NCLS; i += 256)        { float d = b2c[i] - b2[i]; ab += d * d; }
  sw[tid] = aw; sb[tid] = ab;
  __syncthreads();
  for (int s = 128; s; s >>= 1) {
    if (tid < s) { sw[tid] += sw[tid + s]; sb[tid] += sb[tid + s]; }
    __syncthreads();
  }
  if (tid == 0) {
    float nw = sqrtf(sw[0] + 1e-24f);
    float nb = sqrtf(sb[0] + 1e-24f);
    float d = nw + nb + 2e-12f;
    coef[0] = d / (2.f * 10.0f) / nw;   // ALPHA_M = 10
    coef[1] = d / (2.f * 10.0f) / nb;
  }
}

// ============ logits = Msel @ w2c^T + b2c : [16384, NPAD] ============
// One wave per m-tile (1024 waves); sweeps all 7 class n-tiles in registers.
__global__ void k_logits(const float* __restrict__ Msel, const float* __restrict__ w2c,
                         const float* __restrict__ b2c, float* __restrict__ logits) {
  int wave = (blockIdx.x * blockDim.x + threadIdx.x) >> 5;
  int lane = threadIdx.x & 31;
  int mt = wave;
  int half = lane >> 4, lr = lane & 15;
  const float* arow = Msel + (size_t)(mt * 16 + lr) * IDIM;
  const float* bptr[7];
  float bmask[7];
#pragma unroll
  for (int nt = 0; nt < 7; ++nt) {
    int n = nt * 16 + lr;
    int nc = n < NCLS ? n : NCLS - 1;
    bptr[nt] = w2c + (size_t)nc * IDIM;
    bmask[nt] = n < NCLS ? 1.f : 0.f;
  }
  v8f acc[7];
#pragma unroll
  for (int nt = 0; nt < 7; ++nt) acc[nt] = (v8f){0.f,0.f,0.f,0.f,0.f,0.f,0.f,0.f};
  for (int k = 0; k < IDIM; k += 4) {
    v2f a = *(const v2f*)(arow + k + 2 * half);
#pragma unroll
    for (int nt = 0; nt < 7; ++nt) {
      v2f b = *(const v2f*)(bptr[nt] + k + 2 * half);
      b *= bmask[nt];
      acc[nt] = wmma4(a, b, acc[nt]);
    }
  }
#pragma unroll
  for (int nt = 0; nt < 7; ++nt) {
    int n = nt * 16 + lr;
    float bias = n < NCLS ? b2c[n] : 0.f;
#pragma unroll
    for (int r = 0; r < 8; ++r) {
      int m = mt * 16 + r + 8 * half;
      logits[(size_t)m * NPAD + n] = acc[nt][r] + bias;
    }
  }
}

// ============ softmax + dL/dlogits, stored transposed Gt[NPAD][RSEL] ========
__global__ void k_softg(const float* __restrict__ logits, const int* __restrict__ lbl,
                        const float* __restrict__ wsel, float* __restrict__ Gt) {
  int wave = (blockIdx.x * blockDim.x + threadIdx.x) >> 5;
  int lane = threadIdx.x & 31;
  const float* lrow = logits + (size_t)wave * NPAD;
  float v[4];
#pragma unroll
  for (int j = 0; j < 4; ++j) {
    int c = lane + 32 * j;
    v[j] = (c < NCLS) ? lrow[c] : -1e30f;
  }
  float mx = fmaxf(fmaxf(v[0], v[1]), fmaxf(v[2], v[3]));
  for (int m = 16; m; m >>= 1) mx = fmaxf(mx, __shfl_xor(mx, m, 32));
  float s = 0.f;
#pragma unroll
  for (int j = 0; j < 4; ++j) {
    v[j] = expf(v[j] - mx);
    if (lane + 32 * j >= NCLS) v[j] = 0.f;
    s += v[j];
  }
  for (int m = 16; m; m >>= 1) s += __shfl_xor(s, m, 32);
  float inv = 1.f / s;
  int lb = lbl[wave];
  float w = wsel[wave];
#pragma unroll
  for (int j = 0; j < 4; ++j) {
    int c = lane + 32 * j;
    if (c < NPAD) {
      float g = (c < NCLS) ? w * (v[j] * inv - (c == lb ? 1.f : 0.f)) : 0.f;
      Gt[(size_t)c * RSEL + wave] = g;
    }
  }
}

// ============ grad_w2 partials: part[chunk] = (Gt chunk) @ (Msel chunk) =====
// One wave per (chunk, n-tile) = 16 x 32 = 512 waves; sweeps all 7 c-tiles,
// reusing the contiguous MselT B-fragment (b64 loads both operands).
__global__ void k_gradw(const float* __restrict__ Gt, const float* __restrict__ MselT,
                        float* __restrict__ part) {
  int wave = (blockIdx.x * blockDim.x + threadIdx.x) >> 5;
  int lane = threadIdx.x & 31;
  int chunk = wave / 32, nt = wave % 32;
  int half = lane >> 4, lr = lane & 15;
  int n = nt * 16 + lr;
  const float* brow = MselT + (size_t)n * RSEL;
  const float* aptr[7];
#pragma unroll
  for (int mt = 0; mt < 7; ++mt) aptr[mt] = Gt + (size_t)(mt * 16 + lr) * RSEL;
  int k0 = chunk * (RSEL / NCHUNK), k1 = k0 + RSEL / NCHUNK;
  v8f acc[7];
#pragma unroll
  for (int mt = 0; mt < 7; ++mt) acc[mt] = (v8f){0.f,0.f,0.f,0.f,0.f,0.f,0.f,0.f};
  for (int k = k0; k < k1; k += 4) {
    int kk = k + 2 * half;
    v2f b = *(const v2f*)(brow + kk);
#pragma unroll
    for (int mt = 0; mt < 7; ++mt) {
      v2f a = *(const v2f*)(aptr[mt] + kk);
      acc[mt] = wmma4(a, b, acc[mt]);
    }
  }
#pragma unroll
  for (int mt = 0; mt < 7; ++mt) {
#pragma unroll
    for (int r = 0; r < 8; ++r) {
      int c = mt * 16 + r + 8 * half;
      part[((size_t)chunk * NPAD + c) * IDIM + n] = acc[mt][r];
    }
  }
}

// ============ Adam update for w2 (fold partial-reduce + param_dis) ==========
__global__ void k_adamw(const float* __restrict__ part, const float* __restrict__ w2,
                        const float* __restrict__ coef, float* __restrict__ w2c,
                        float* __restrict__ mw, float* __restrict__ vw,
                        float bc1inv, float bc2inv) {
  int i = blockIdx.x * blockDim.x + threadIdx.x;
  if (i >= NCLS * IDIM) return;
  int c = i / IDIM, d = i % IDIM;
  float g = 0.f;
#pragma unroll
  for (int ch = 0; ch < NCHUNK; ++ch)
    g += part[((size_t)ch * NPAD + c) * IDIM + d];
  g += coef[0] * (w2c[i] - w2[i]);
  float m = 0.9f * mw[i] + 0.1f * g;
  float v = 0.999f * vw[i] + 0.001f * g * g;
  mw[i] = m; vw[i] = v;
  w2c[i] -= 1e-3f * (m * bc1inv) / (sqrtf(v * bc2inv) + 1e-8f);
}

// ============ Adam update for b2 (one wave per class) ============
__global__ void k_adamb(const float* __restrict__ Gt, const float* __restrict__ b2,
                        const float* __restrict__ coef, float* __restrict__ b2c,
                        float* __restrict__ mb, float* __restrict__ vb,
                        float bc1inv, float bc2inv) {
  int wave = (blockIdx.x * blockDim.x + threadIdx.x) >> 5;
  int lane = threadIdx.x & 31;
  if (wave >= NCLS) return;
  const float* row = Gt + (size_t)wave * RSEL;
  float s = 0.f;
  for (int i = lane * 4; i < RSEL; i += 128) {
    float4 x = *(const float4*)(row + i);
    s += x.x + x.y + x.z + x.w;
  }
  for (int m = 16; m; m >>= 1) s += __shfl_xor(s, m, 32);
  if (lane == 0) {
    float g = s + coef[1] * (b2c[wave] - b2[wave]);
    float m = 0.9f * mb[wave] + 0.1f * g;
    float v = 0.999f * vb[wave] + 0.001f * g * g;
    mb[wave] = m; vb[wave] = v;
    b2c[wave] -= 1e-3f * (m * bc1inv) / (sqrtf(v * bc2inv) + 1e-8f);
  }
}

// ============ out = emb @ w2c^T + b2c : [128, 100] ============
__global__ void k_final(const float* __restrict__ emb, const float* __restrict__ w2c,
                        const float* __restrict__ b2c, float* __restrict__ out) {
  int i = blockIdx.x * blockDim.x + threadIdx.x;
  if (i >= BSZ * NCLS) return;
  int b = i / NCLS, c = i % NCLS;
  const float* e = emb + (size_t)b * IDIM;
  const float* w = w2c + (size_t)c * IDIM;
  float s = 0.f;
#pragma unroll 4
  for (int d = 0; d < IDIM; d += 4) {
    float4 ev = *(const float4*)(e + d);
    float4 wv = *(const float4*)(w + d);
    s += ev.x * wv.x + ev.y * wv.y + ev.z * wv.z + ev.w * wv.w;
  }
  out[i] = s + b2c[c];
}

extern "C" void kernel_launch(void* const* d_in, const int* in_sizes, int n_in,
                              void* d_out, int out_size, void* d_ws, size_t ws_size,
                              hipStream_t stream) {
  const float* input = (const float*)d_in[0];
  const float* memx  = (const float*)d_in[1];
  const float* w1    = (const float*)d_in[2];
  const float* b1    = (const float*)d_in[3];
  const float* w2    = (const float*)d_in[4];
  const float* b2    = (const float*)d_in[5];
  const int*   memy  = (const int*)d_in[6];
  float* out = (float*)d_out;

  float* p = (float*)d_ws;
  float* emb  = p; p += BSZ * IDIM;
  float* embn = p; p += BSZ;
  float* memn = p; p += CAP;
  float* w2c  = p; p += NCLS * IDIM;
  float* mw   = p; p += NCLS * IDIM;
  float* vw   = p; p += NCLS * IDIM;
  float* b2c  = p; p += 128;
  float* mb   = p; p += 128;
  float* vb   = p; p += 128;
  float* coef = p; p += 8;
  float* wsel = p; p += RSEL;
  int*   idx  = (int*)p; p += RSEL;
  int*   lbl  = (int*)p; p += RSEL;
  float* Msel  = p; p += (size_t)RSEL * IDIM;
  float* MselT = p; p += (size_t)RSEL * IDIM;
  float* logits = p; p += (size_t)RSEL * NPAD;
  float* Gt   = p; p += (size_t)NPAD * RSEL;
  float* part = p; p += (size_t)NCHUNK * NPAD * IDIM;
  float* kernT = p; p += (size_t)BSZ * CAP;

  // Phase 1: embedding, norms, distances, top-k, gather
  k_emb<<<32, 256, 0, stream>>>(input, w1, b1, emb);
  k_rownorm<<<BSZ / 8, 256, 0, stream>>>(emb, embn, BSZ);
  k_rownorm<<<CAP / 8, 256, 0, stream>>>(memx, memn, CAP);
  k_dist<<<(CAP / 16 + 7) / 8, 256, 0, stream>>>(memx, emb, memn, embn, kernT);
  k_topk<<<BSZ, 512, 0, stream>>>(kernT, idx, wsel);
  k_gather<<<RSEL / 8, 256, 0, stream>>>(memx, memy, idx, Msel, MselT, lbl);
  k_initp<<<200, 256, 0, stream>>>(w2, b2, w2c, mw, vw, b2c, mb, vb);

  // Phase 2: 41 Adam steps on (w2, b2) only — w1/b1 gradients are exactly 0
  double B1t = 1.0, B2t = 1.0;
  for (int t = 1; t <= STEPS; ++t) {
    B1t *= 0.9; B2t *= 0.999;
    float bc1inv = (float)(1.0 / (1.0 - B1t));
    float bc2inv = (float)(1.0 / (1.0 - B2t));
    k_norms<<<1, 256, 0, stream>>>(w2c, w2, b2c, b2, coef);
    k_logits<<<RSEL / 16 / 8, 256, 0, stream>>>(Msel, w2c, b2c, logits);
    k_softg<<<RSEL / 8, 256, 0, stream>>>(logits, lbl, wsel, Gt);
    k_gradw<<<64, 256, 0, stream>>>(Gt, MselT, part);
    k_adamw<<<200, 256, 0, stream>>>(part, w2, coef, w2c, mw, vw, bc1inv, bc2inv);
    k_adamb<<<13, 256, 0, stream>>>(Gt, b2, coef, b2c, mb, vb, bc1inv, bc2inv);
  }

  // Phase 3: out = relu(input@w1^T+b1) @ w2^T + b2  (hid == emb, w1 frozen)
  k_final<<<50, 256, 0, stream>>>(emb, w2c, b2c, out);
}